// SRNN_2RNN_general_90417651516155
// MI455X (gfx1250) — compile-verified
//
#include <hip/hip_runtime.h>
#include <cmath>

typedef float v2f __attribute__((ext_vector_type(2)));
typedef float v8f __attribute__((ext_vector_type(8)));

#define HID     256
#define STRIDE  260        // LDS row stride (floats), breaks bank conflicts
#define TSTEPS  784
#define MT      16         // batch rows per workgroup (WMMA M tile)
#define BLOCK   512        // 16 wave32s; each wave owns one 16-wide N tile

static __device__ __forceinline__ v8f wmma4(v2f a, v2f b, v8f c) {
  // V_WMMA_F32_16X16X4_F32: exact fp32 D = A(16x4) * B(4x16) + C
  return __builtin_amdgcn_wmma_f32_16x16x4_f32(false, a, false, b, (short)0, c,
                                               false, false);
}

__global__ __launch_bounds__(BLOCK) void srnn_kernel(
    const float* __restrict__ x,
    const float* __restrict__ w_i2h1, const float* __restrict__ b_i2h1,
    const float* __restrict__ w_h2h1, const float* __restrict__ b_h2h1,
    const float* __restrict__ w_i2h2, const float* __restrict__ b_i2h2,
    const float* __restrict__ w_h2h2, const float* __restrict__ b_h2h2,
    const float* __restrict__ w_i2h3, const float* __restrict__ b_i2h3,
    const float* __restrict__ w_h2o3, const float* __restrict__ b_h2o3,
    float* __restrict__ out)
{
  __shared__ float spk1_lds[MT * STRIDE];
  __shared__ float spk2_lds[MT * STRIDE];
  __shared__ float x_lds[MT];

  const int tid   = threadIdx.x;
  const int wave  = tid >> 5;            // 0..15
  const int lane  = tid & 31;
  const int half  = lane >> 4;           // 0/1: K (operand) / M (C-tile) halves
  const int ml    = lane & 15;
  const int mbase = blockIdx.x * MT;
  const int n     = wave * MT + ml;      // hidden column owned by this lane

  for (int i = tid; i < MT * STRIDE; i += BLOCK) {
    spk1_lds[i] = 0.0f; spk2_lds[i] = 0.0f;
  }
  if (tid < MT) x_lds[tid] = x[(size_t)(mbase + tid) * TSTEPS];

  // Per-column constants (biases folded, layer-1 input weight)
  const float bias1 = b_i2h1[n] + b_h2h1[n];
  const float bias2 = b_i2h2[n] + b_h2h2[n];
  const float bias3 = b_i2h3[n];
  const float w1r   = w_i2h1[n];

  // Mask parameters: replicate numpy linspace/ceil/rint in double.
  // MASK1==MASK2==MASK3 (identical make_mask params).
  int cyc, onl, ph;
  {
    double c  = (double)n * (4.0 / 255.0) + 4.0;
    cyc = (int)ceil(c); if (cyc > 8) cyc = 8;          // endpoint exact in numpy
    double dc = (double)n * (0.8 / 255.0) + 0.1;
    onl = (int)ceil(dc * (double)cyc);
    int ps = (int)rint((double)n * (4.0 / 255.0));     // no exact .5 cases
    ph = (-ps) % cyc; if (ph < 0) ph += cyc;           // phase of (t - ps) mod cyc
  }

  const v8f vzero = {};
  v8f mem1 = vzero, mem2 = vzero, mem3 = vzero;
  v8f spk1 = vzero, spk2 = vzero, spk3 = vzero, cnt = vzero;

  // A operand: ds_load_b64 at spk[m][kb + 2*half] -> (K=2h, K=2h+1) per A layout
  const int aoff = ml * STRIDE + 2 * half;
  // B operand: global_load_b64 at W[n][kb + 2*half] (B mirrors A with lane=N)
  const float* __restrict__ bp_h2h1 = w_h2h1 + (size_t)n * HID + 2 * half;
  const float* __restrict__ bp_i2h2 = w_i2h2 + (size_t)n * HID + 2 * half;
  const float* __restrict__ bp_h2h2 = w_h2h2 + (size_t)n * HID + 2 * half;
  const float* __restrict__ bp_i2h3 = w_i2h3 + (size_t)n * HID + 2 * half;

  __syncthreads();

  for (int t = 0; t < TSTEPS; ++t) {
    const float mk = (ph < onl) ? 1.0f : 0.0f;

    // ---------- layer 1: h1 = spk1_prev @ w_h2h1^T + x_t*w_i2h1 + biases ----
    v8f a0 = vzero, a1 = vzero;   // two independent K-chains (k<128, k>=128)
    {
      const float* ap = spk1_lds + aoff;
      #pragma unroll 8
      for (int kc = 0; kc < 32; ++kc) {
        v2f aL = *(const v2f*)(ap + kc * 4);
        v2f aH = *(const v2f*)(ap + kc * 4 + 128);
        v2f bL = *(const v2f*)(bp_h2h1 + kc * 4);
        v2f bH = *(const v2f*)(bp_h2h1 + kc * 4 + 128);
        a0 = wmma4(aL, bL, a0);
        a1 = wmma4(aH, bH, a1);
      }
      // rank-1 x-term as one extra WMMA chunk: only K-slot (vgpr0, half0) live
      v2f ax; ax.x = half ? 0.0f : x_lds[ml]; ax.y = 0.0f;
      v2f bx; bx.x = half ? 0.0f : w1r;       bx.y = 0.0f;
      a0 = wmma4(ax, bx, a0);
    }
    __syncthreads();                       // all layer-1 reads done
    #pragma unroll
    for (int r = 0; r < 8; ++r) {          // lane owns (m = half*8+r, col n)
      float hval = a0[r] + a1[r] + bias1;
      float pre  = mem1[r];
      float nm   = pre * 0.5f * (1.0f - spk1[r]) + hval;
      float mnew = (mk != 0.0f) ? nm : pre;
      mem1[r] = mnew;
      float s = (mnew > 0.5f) ? mk : 0.0f;
      spk1[r] = s;
      spk1_lds[(half * 8 + r) * STRIDE + n] = s;
    }
    if (tid < MT && t + 1 < TSTEPS)        // stage next x column
      x_lds[tid] = x[(size_t)(mbase + tid) * TSTEPS + t + 1];
    __syncthreads();                       // spk1/x updated

    // ---------- layer 2: h2 = spk1 @ w_i2h2^T + spk2_prev @ w_h2h2^T + biases
    a0 = vzero; a1 = vzero;
    {
      const float* ap1 = spk1_lds + aoff;
      const float* ap2 = spk2_lds + aoff;
      #pragma unroll 8
      for (int kc = 0; kc < 32; ++kc) {
        v2f aL = *(const v2f*)(ap1 + kc * 4);
        v2f aH = *(const v2f*)(ap1 + kc * 4 + 128);
        v2f bL = *(const v2f*)(bp_i2h2 + kc * 4);
        v2f bH = *(const v2f*)(bp_i2h2 + kc * 4 + 128);
        a0 = wmma4(aL, bL, a0);
        a1 = wmma4(aH, bH, a1);
      }
      #pragma unroll 8
      for (int kc = 0; kc < 32; ++kc) {
        v2f aL = *(const v2f*)(ap2 + kc * 4);
        v2f aH = *(const v2f*)(ap2 + kc * 4 + 128);
        v2f bL = *(const v2f*)(bp_h2h2 + kc * 4);
        v2f bH = *(const v2f*)(bp_h2h2 + kc * 4 + 128);
        a0 = wmma4(aL, bL, a0);
        a1 = wmma4(aH, bH, a1);
      }
    }
    __syncthreads();                       // layer-2 reads of spk2 done
    #pragma unroll
    for (int r = 0; r < 8; ++r) {
      float hval = a0[r] + a1[r] + bias2;
      float pre  = mem2[r];
      float nm   = pre * 0.5f * (1.0f - spk2[r]) + hval;
      float mnew = (mk != 0.0f) ? nm : pre;
      mem2[r] = mnew;
      float s = (mnew > 0.5f) ? mk : 0.0f;
      spk2[r] = s;
      spk2_lds[(half * 8 + r) * STRIDE + n] = s;
    }
    __syncthreads();                       // spk2 updated

    // ---------- layer 3: h3 = spk2 @ w_i2h3^T + b; accumulate spike counts --
    a0 = vzero; a1 = vzero;
    {
      const float* ap = spk2_lds + aoff;
      #pragma unroll 8
      for (int kc = 0; kc < 32; ++kc) {
        v2f aL = *(const v2f*)(ap + kc * 4);
        v2f aH = *(const v2f*)(ap + kc * 4 + 128);
        v2f bL = *(const v2f*)(bp_i2h3 + kc * 4);
        v2f bH = *(const v2f*)(bp_i2h3 + kc * 4 + 128);
        a0 = wmma4(aL, bL, a0);
        a1 = wmma4(aH, bH, a1);
      }
    }
    #pragma unroll
    for (int r = 0; r < 8; ++r) {
      float hval = a0[r] + a1[r] + bias3;
      float pre  = mem3[r];
      float nm   = pre * 0.5f * (1.0f - spk3[r]) + hval;
      float mnew = (mk != 0.0f) ? nm : pre;
      mem3[r] = mnew;
      float s = (mnew > 0.5f) ? mk : 0.0f;
      spk3[r] = s;
      cnt[r] += s;                         // out GEMM is linear: count spikes
    }
    ph++; if (ph == cyc) ph = 0;
  }

  // ---------- output: out = (cnt @ w_h2o3^T + T*b) / T -----------------------
  __syncthreads();
  #pragma unroll
  for (int r = 0; r < 8; ++r)
    spk1_lds[(half * 8 + r) * STRIDE + n] = cnt[r];   // reuse buffer
  __syncthreads();

  if (tid < MT * 10) {
    const int m = tid / 10, o = tid % 10;
    const float* wo = w_h2o3 + (size_t)o * HID;
    const float* cl = spk1_lds + m * STRIDE;
    float s = 0.0f;
    for (int hh = 0; hh < HID; ++hh) s += cl[hh] * wo[hh];
    out[(size_t)(mbase + m) * 10 + o] =
        (s + (float)TSTEPS * b_h2o3[o]) / (float)TSTEPS;
  }
}

extern "C" void kernel_launch(void* const* d_in, const int* in_sizes, int n_in,
                              void* d_out, int out_size, void* d_ws, size_t ws_size,
                              hipStream_t stream) {
  const float* x      = (const float*)d_in[0];
  const float* w_i2h1 = (const float*)d_in[1];
  const float* b_i2h1 = (const float*)d_in[2];
  const float* w_h2h1 = (const float*)d_in[3];
  const float* b_h2h1 = (const float*)d_in[4];
  const float* w_i2h2 = (const float*)d_in[5];
  const float* b_i2h2 = (const float*)d_in[6];
  const float* w_h2h2 = (const float*)d_in[7];
  const float* b_h2h2 = (const float*)d_in[8];
  const float* w_i2h3 = (const float*)d_in[9];
  const float* b_i2h3 = (const float*)d_in[10];
  const float* w_h2o3 = (const float*)d_in[11];
  const float* b_h2o3 = (const float*)d_in[12];

  const int N      = in_sizes[0] / TSTEPS;   // 512
  const int blocks = N / MT;                 // 32 persistent workgroups

  srnn_kernel<<<blocks, BLOCK, 0, stream>>>(
      x, w_i2h1, b_i2h1, w_h2h1, b_h2h1, w_i2h2, b_i2h2, w_h2h2, b_h2h2,
      w_i2h3, b_i2h3, w_h2o3, b_h2o3, (float*)d_out);
}